// ScaledDotProductAttention_18124761989668
// MI455X (gfx1250) — compile-verified
//
#include <hip/hip_runtime.h>

typedef float v2f __attribute__((ext_vector_type(2)));
typedef float v4f __attribute__((ext_vector_type(4)));
typedef float v8f __attribute__((ext_vector_type(8)));
typedef int   v4i __attribute__((ext_vector_type(4)));

#define SEQ    8192
#define DIM    64
#define NQ     4               // column quarters per strip (1 wave each)
#define QCOLS  (SEQ / NQ)      // 2048 columns per wave
#define STEPS  (QCOLS / 16)    // 128 16-wide tiles per wave per pass
#define RSTR   68              // padded LDS row stride (floats): banks 4n+2h+4j -> conflict-free
#define TILE_F (16 * RSTR)     // floats per staged 16x64 K tile (padded)
#define QSCALE 0.18033688011112042f   // log2(e) / TEMPERATURE(=8): scores live in exp2 domain
#define EPS_F  1e-7f

#define AS1 __attribute__((address_space(1)))
#define AS3 __attribute__((address_space(3)))

#if __has_builtin(__builtin_amdgcn_global_load_async_to_lds_b128)
#define HAVE_ASYNC 1
#else
#define HAVE_ASYNC 0
#endif

static __device__ __forceinline__ float ex2(float x) { return __builtin_amdgcn_exp2f(x); }

template <int N>
static __device__ __forceinline__ void wait_async() {
#if HAVE_ASYNC
#if __has_builtin(__builtin_amdgcn_s_wait_asynccnt)
    __builtin_amdgcn_s_wait_asynccnt(N);
#else
    asm volatile("s_wait_asynccnt %0" :: "i"(N) : "memory");
#endif
#endif
}

// Copy one 16B chunk global -> LDS (async DMA path, ASYNCcnt-tracked).
static __device__ __forceinline__ void copy16(const float* __restrict__ g, float* l) {
#if HAVE_ASYNC
    __builtin_amdgcn_global_load_async_to_lds_b128(
        (AS1 v4i*)(unsigned long long)(uintptr_t)g,
        (AS3 v4i*)(unsigned)(uintptr_t)l,     // flat LDS addr low 32 bits == LDS offset
        0, 0);
#else
    *(v4f*)l = *(const v4f*)g;
#endif
}

// Stage the 4 quarters' K tiles for `step` into buffer `dbuf`.
// 4 tiles x 16 rows x 16 chunks(16B) = 1024 chunks; 512 threads x 2 chunks.
static __device__ __forceinline__ void stage_tiles(const float* __restrict__ K,
                                                   float* __restrict__ lds,
                                                   int step, int dbuf, int tid) {
#pragma unroll
    for (int c0 = 0; c0 < 2; ++c0) {
        int c   = tid + c0 * 512;
        int qq  = c >> 8;          // quarter 0..3
        int cc  = c & 255;
        int row = cc >> 4;         // key row within tile 0..15
        int k4  = cc & 15;         // 16B chunk within row
        const float* g = K + (size_t)(qq * QCOLS + step * 16 + row) * DIM + k4 * 4;
        float* l = lds + (size_t)(dbuf * NQ + qq) * TILE_F + row * RSTR + k4 * 4;
        copy16(g, l);
    }
}

// One 16(M) x 16(N) score tile; K-dim = 64 as 16 chained V_WMMA_F32_16X16X4_F32.
// kb points into LDS at (n-row nlo, d-offset 2h) of the staged tile; chunk j at +4j.
static __device__ __forceinline__ v8f score_tile(const v2f a[16], const float* kb) {
    v8f c = {0.f, 0.f, 0.f, 0.f, 0.f, 0.f, 0.f, 0.f};
#pragma unroll
    for (int j = 0; j < 16; ++j) {
        v2f b = *(const v2f*)(kb + 4 * j);   // ds_load_b64, bank-conflict-free (RSTR=68)
        c = __builtin_amdgcn_wmma_f32_16x16x4_f32(false, a[j], false, b,
                                                  (short)0, c, false, false);
    }
    return c;
}

// Branchless sorted insert of s into descending 6-list.
static __device__ __forceinline__ void ins6(float t[6], float s) {
#pragma unroll
    for (int j = 0; j < 6; ++j) {
        float mx = fmaxf(t[j], s);
        s = fminf(t[j], s);
        t[j] = mx;
    }
}

__global__ __launch_bounds__(512)
void topk_attn_kernel(const float* __restrict__ Q, const float* __restrict__ K,
                      float* __restrict__ Out) {
    __shared__ float kstage[2 * NQ * TILE_F];   // double-buffered staged K tiles (~34 KB)
    __shared__ float sbuf[4][NQ][16][8];        // per (strip, quarter, row): top6 + Z
    __shared__ float cbuf[4][16][4];            // per (strip, row): v5, e6, invD

    const int tid   = threadIdx.x;
    const int lane  = tid & 31;
    const int wv    = tid >> 5;       // 0..15
    const int sl    = wv >> 2;        // strip-local 0..3
    const int q     = wv & 3;         // column quarter 0..3
    const int strip = blockIdx.x * 4 + sl;     // 0..511
    const int m0    = strip * 16;
    const int nlo   = lane & 15;
    const int h     = lane >> 4;

    // ---- A fragment: Q rows m0..m0+15, scale (temperature + log2e) folded in ----
    v2f a[16];
    {
        const float* qp = Q + (size_t)(m0 + nlo) * DIM + 2 * h;
#pragma unroll
        for (int j = 0; j < 16; ++j) {
            v2f t = *(const v2f*)(qp + 4 * j);
            a[j].x = t.x * QSCALE;
            a[j].y = t.y * QSCALE;
        }
    }

    // ---- Pass 1: per-lane running sum-of-exp + branchless top-6 over this quarter ----
    float top[8][6], Z[8];
#pragma unroll
    for (int r = 0; r < 8; ++r) {
        Z[r] = 0.f;
#pragma unroll
        for (int j = 0; j < 6; ++j) top[r][j] = -3.0e38f;
    }

    stage_tiles(K, kstage, 0, 0, tid);
    for (int i = 0; i < STEPS; ++i) {
        if (i + 1 < STEPS) {
            stage_tiles(K, kstage, i + 1, (i + 1) & 1, tid);
            wait_async<2>();            // step i's copies done; step i+1 in flight
        } else {
            wait_async<0>();
        }
        __syncthreads();                // staged tiles visible to all waves
        const float* kb = kstage + (size_t)(((i & 1) * NQ + q)) * TILE_F + nlo * RSTR + 2 * h;
        v8f c = score_tile(a, kb);
#pragma unroll
        for (int r = 0; r < 8; ++r) {
            float s = c[r];
            Z[r] += ex2(s);
            ins6(top[r], s);
        }
        __syncthreads();                // all reads done before buffer reuse
    }

    // ---- Cross-lane reduce within each 16-lane half (row r+8h lives in half h) ----
#pragma unroll
    for (int msk = 1; msk <= 8; msk <<= 1) {
#pragma unroll
        for (int r = 0; r < 8; ++r) {
            Z[r] += __shfl_xor(Z[r], msk, 32);
            float b[6];
#pragma unroll
            for (int j = 0; j < 6; ++j) b[j] = __shfl_xor(top[r][j], msk, 32);
#pragma unroll
            for (int j = 0; j < 6; ++j) ins6(top[r], b[j]);
        }
    }

    // ---- Publish per-quarter stats, merge quarters, derive row constants ----
    if (nlo == 0) {
#pragma unroll
        for (int r = 0; r < 8; ++r) {
            int row = r + 8 * h;
#pragma unroll
            for (int j = 0; j < 6; ++j) sbuf[sl][q][row][j] = top[r][j];
            sbuf[sl][q][row][6] = Z[r];
        }
    }
    __syncthreads();
    if (tid < 64) {                     // one thread per (strip, row)
        int s2 = tid >> 4, row = tid & 15;
        float t6[6];
#pragma unroll
        for (int j = 0; j < 6; ++j) t6[j] = sbuf[s2][0][row][j];
        float Zt = sbuf[s2][0][row][6];
#pragma unroll
        for (int qq = 1; qq < NQ; ++qq) {
            Zt += sbuf[s2][qq][row][6];
#pragma unroll
            for (int j = 0; j < 6; ++j) ins6(t6, sbuf[s2][qq][row][j]);
        }
        // out_i = (e_i - e6) / ((S6 - 6*e6) + eps*Z)  for s_i > v5, else 0
        float v5 = t6[5];
        float e6 = ex2(v5);
        float S6 = 0.f;
#pragma unroll
        for (int j = 0; j < 6; ++j) S6 += ex2(t6[j]);
        cbuf[s2][row][0] = v5;
        cbuf[s2][row][1] = e6;
        cbuf[s2][row][2] = 1.0f / ((S6 - 6.0f * e6) + EPS_F * Zt);
    }
    __syncthreads();

    float v5[8], e6[8], invD[8];
#pragma unroll
    for (int r = 0; r < 8; ++r) {
        int row = r + 8 * h;
        v5[r]   = cbuf[sl][row][0];
        e6[r]   = cbuf[sl][row][1];
        invD[r] = cbuf[sl][row][2];
    }

    // ---- Pass 2: recompute tiles, stream dense thresholded output (NT stores) ----
    float* ob = Out + (size_t)(m0 + 8 * h) * SEQ + q * QCOLS + nlo;
    stage_tiles(K, kstage, 0, 0, tid);
    for (int i = 0; i < STEPS; ++i, ob += 16) {
        if (i + 1 < STEPS) {
            stage_tiles(K, kstage, i + 1, (i + 1) & 1, tid);
            wait_async<2>();
        } else {
            wait_async<0>();
        }
        __syncthreads();
        const float* kb = kstage + (size_t)(((i & 1) * NQ + q)) * TILE_F + nlo * RSTR + 2 * h;
        v8f c = score_tile(a, kb);
#pragma unroll
        for (int r = 0; r < 8; ++r) {
            float s = c[r];
            float e = ex2(s);
            float o = (s > v5[r]) ? (e - e6[r]) * invD[r] : 0.0f;
            __builtin_nontemporal_store(o, ob + (size_t)r * SEQ);  // write-once 256MB stream
        }
        __syncthreads();
    }
}

extern "C" void kernel_launch(void* const* d_in, const int* in_sizes, int n_in,
                              void* d_out, int out_size, void* d_ws, size_t ws_size,
                              hipStream_t stream) {
    (void)in_sizes; (void)n_in; (void)d_ws; (void)ws_size; (void)out_size;
    const float* q = (const float*)d_in[0];   // [1, 8192, 64] f32
    const float* k = (const float*)d_in[1];   // [1, 8192, 64] f32
    float* out = (float*)d_out;               // [8192, 8192] f32
    // 128 blocks x 512 threads: each block = 4 strips x 4 column-quarters (16 waves),
    // cooperatively staging K tiles through LDS; 2048 waves total (~4/SIMD).
    topk_attn_kernel<<<dim3(128), dim3(512), 0, stream>>>(q, k, out);
}